// RipsLayer_25769803776473
// MI455X (gfx1250) — compile-verified
//
#include <hip/hip_runtime.h>
#include <stdint.h>

typedef unsigned int u32;
typedef unsigned long long u64;
typedef __attribute__((ext_vector_type(4))) u32 v4u;
typedef __attribute__((ext_vector_type(8))) int  v8i;
typedef __attribute__((ext_vector_type(4))) int  v4i;

#define NPTS 8192
#define NDIM 3
#define NFLT (NPTS * NDIM)   // 24576 floats = 96 KB, fits in 320KB WGP LDS

// dgm[e] = || X[i_e] - X[j_e] ||  for flattened entries e in [0, 2P)
// indices flat layout [P,2,2] -> i = idx[2e], j = idx[2e+1]
__global__ __launch_bounds__(256) void rips_pairs_kernel(
    const float* __restrict__ X,
    const int*   __restrict__ idx,
    float*       __restrict__ out,
    int n_entries)
{
    __shared__ __align__(16) float sX[NFLT];

#if __has_builtin(__builtin_amdgcn_tensor_load_to_lds) && __has_builtin(__builtin_amdgcn_s_wait_tensorcnt)
    // --- Stage all of X (96 KB) into LDS with one TDM descriptor (wave 0 issues) ---
    if (threadIdx.x < 32) {
        // LDS byte offset: flat address low 32 bits map directly to LDS (aperture rule)
        u32 lds_off = (u32)(uintptr_t)&sX[0];
        u64 ga      = (u64)(uintptr_t)X;

        // D# Group 0 (128b): count=1(valid), is_restore=0, gather_mode=0,
        // lds_addr[63:32], global_addr[120:64], type=2 at [127:126]
        v4u g0;
        g0[0] = 1u;
        g0[1] = lds_off;
        g0[2] = (u32)ga;
        g0[3] = ((u32)(ga >> 32) & 0x01FFFFFFu) | (2u << 30);

        // D# Group 1 (256b): workgroup_mask=0 (not in cluster), data_size=2 (4B),
        // no atomic-barrier / iterate / pad. 1-D tile: tensor_dim0 = tile_dim0 = 24576,
        // tensor_dim1 = 1, tile_dim1/2 = 0 (unused).
        v8i g1;
        g1[0] = (int)(2u << 16);                         // data_size = 4 bytes
        g1[1] = (int)((u32)(NFLT & 0xFFFF) << 16);       // tensor_dim0 lo16 @ [63:48]
        g1[2] = (int)(((u32)NFLT >> 16) | (1u << 16));   // tensor_dim0 hi16 | tensor_dim1=1
        g1[3] = (int)((u32)(NFLT & 0xFFFF) << 16);       // tile_dim0 @ [127:112]
        g1[4] = 0;                                       // tile_dim1=0, tile_dim2=0
        g1[5] = (int)NFLT;                               // tensor_dim0_stride lo32
        g1[6] = 0;
        g1[7] = 0;

        v4i gz4 = {0, 0, 0, 0};                          // groups 2/3: dims unused (1-D tile)
        v8i gz8 = {0, 0, 0, 0, 0, 0, 0, 0};              // extra group (6-arg toolchain form)
        __builtin_amdgcn_tensor_load_to_lds(g0, g1, gz4, gz4, gz8, /*cpol=*/0);
        __builtin_amdgcn_s_wait_tensorcnt(0);            // TENSORcnt == 0 -> tile in LDS
    }
#else
    // Fallback staging path (keeps the kernel compiling if the TDM builtin is absent)
    for (int t = threadIdx.x; t < NFLT; t += blockDim.x) sX[t] = X[t];
#endif
    __syncthreads();
    // TDM wrote LDS invisibly to the compiler; fence its view of memory.
    asm volatile("" ::: "memory");

    // --- Gather phase: random (i,j) pairs resolved from LDS ---
    const int stride = (int)(gridDim.x * blockDim.x);
    for (int e = (int)(blockIdx.x * blockDim.x + threadIdx.x); e < n_entries; e += stride) {
        const int2 ij = ((const int2*)idx)[e];           // one global_load_b64
        const float* a = &sX[3 * ij.x];
        const float* b = &sX[3 * ij.y];
        float dx = a[0] - b[0];
        float dy = a[1] - b[1];
        float dz = a[2] - b[2];
        out[e] = sqrtf(dx * dx + dy * dy + dz * dz);
    }
}

extern "C" void kernel_launch(void* const* d_in, const int* in_sizes, int n_in,
                              void* d_out, int out_size, void* d_ws, size_t ws_size,
                              hipStream_t stream) {
    const float* X   = (const float*)d_in[0];   // [N, 3] fp32
    const int*   idx = (const int*)d_in[1];     // [P, 2, 2] int32
    float*       out = (float*)d_out;           // [P, 2] fp32 (flat)

    const int n_entries = out_size;             // = 2 * P = in_sizes[1] / 2

    // 32 WGs x 256 threads: each WG stages X once via TDM (96 KB -> 3 MB total L2
    // reads across WGs, negligible), then grid-stride over the 16K gathers.
    rips_pairs_kernel<<<32, 256, 0, stream>>>(X, idx, out, n_entries);
}